// MDDecoder_89653147337380
// MI455X (gfx1250) — compile-verified
//
#include <hip/hip_runtime.h>
#include <hip/hip_bf16.h>

// MI455X / gfx1250: wave32, WMMA. Poincare-ball pairwise sqdist + NCE-style loss.
// Pipeline: init(norms, x-copy, ws zero) -> gram tiles (V_WMMA_F32_16X16X4_F32 x4)
// fused with sqdist/exp + row neg_sum + global max/min partials -> loss pass ->
// finalize scalars.  Memory-bound (~256MB total traffic @ 23.3 TB/s); GEMM is
// negligible so full-precision f32 WMMA is the correct choice.

typedef float v2f __attribute__((ext_vector_type(2)));
typedef float v8f __attribute__((ext_vector_type(8)));

#define NPTS 4096
#define DIMK 16
#define EPSF 1e-15f
#define CLIPF (1.0f - 1e-15f)

// ws layout (floats): [0,4096) norms | [4096,8192) neg_sum | [8192..8195] loss,dmax,maxI,minI
#define W_NORM 0
#define W_NEG  NPTS
#define W_CELL (2 * NPTS)

__device__ __forceinline__ void atomicMaxF(float* addr, float val) {
    unsigned int* ua = (unsigned int*)addr;
    unsigned int old = *ua;
    while (val > __uint_as_float(old)) {
        unsigned int assumed = old;
        old = atomicCAS(ua, assumed, __float_as_uint(val));
        if (old == assumed) break;
    }
}
__device__ __forceinline__ void atomicMinF(float* addr, float val) {
    unsigned int* ua = (unsigned int*)addr;
    unsigned int old = *ua;
    while (val < __uint_as_float(old)) {
        unsigned int assumed = old;
        old = atomicCAS(ua, assumed, __float_as_uint(val));
        if (old == assumed) break;
    }
}

// Kernel 0: copy x to output, compute row norms, zero neg_sum, init scalar cells.
__global__ void init_kernel(const float* __restrict__ x,
                            float* __restrict__ out_x,
                            float* __restrict__ w) {
    const int i = blockIdx.x * blockDim.x + threadIdx.x;
    if (i < NPTS) {
        float nrm = 0.f;
#pragma unroll
        for (int k = 0; k < DIMK; k += 4) {
            float4 v = *(const float4*)(x + i * DIMK + k);
            *(float4*)(out_x + i * DIMK + k) = v;
            nrm += v.x * v.x + v.y * v.y + v.z * v.z + v.w * v.w;
        }
        w[W_NORM + i] = nrm;
        w[W_NEG + i] = 0.f;
    }
    if (i == 0) {
        w[W_CELL + 0] = 0.f;                 // loss accumulator
        w[W_CELL + 1] = -__builtin_inff();   // dist.max
        w[W_CELL + 2] = -__builtin_inff();   // max_inner
        w[W_CELL + 3] = __builtin_inff();    // min_inner
    }
}

// Kernel 1: one 16x16 Gram tile per wave via 4x V_WMMA_F32_16X16X4_F32, fused
// with the hyperbolic sqdist math, dist store, neg_sum and max/min partials.
// Block = 256 threads = 8 waves covering 16 rows x 128 cols.
__global__ void __launch_bounds__(256)
gram_tile_kernel(const float* __restrict__ x,
                 const int* __restrict__ adj,
                 float* __restrict__ dist_out,
                 float* __restrict__ w) {
    __shared__ float s_neg[16];
    __shared__ float s_dmax[256];
    __shared__ float s_maxi[256];
    __shared__ float s_mini[256];

    const int lane = threadIdx.x & 31;
    const int wave = threadIdx.x >> 5;
    const int i0 = blockIdx.y * 16;                 // tile row base
    const int j0 = blockIdx.x * 128 + wave * 16;    // tile col base

    if (threadIdx.x < 16) s_neg[threadIdx.x] = 0.f;
    __syncthreads();

    // A(16x4 f32) layout: lanes 0-15 -> M=lane, {K=0,K=1}; lanes 16-31 -> {K=2,K=3}.
    // B(4x16 f32) mirrors it with N = lane&15.
    const int m  = lane & 15;
    const int kh = (lane >> 4) << 1;   // 0 or 2

    v8f c = {};
#pragma unroll
    for (int kk = 0; kk < 4; ++kk) {   // K = 16 in chunks of 4
        v2f a = *(const v2f*)(x + (i0 + m) * DIMK + 4 * kk + kh);
        v2f b = *(const v2f*)(x + (j0 + m) * DIMK + 4 * kk + kh);
        // 8 args: (neg_a, A, neg_b, B, c_mod, C, reuse_a, reuse_b)
        c = __builtin_amdgcn_wmma_f32_16x16x4_f32(false, a, false, b,
                                                  (short)0, c, false, false);
    }

    const float* norms  = w + W_NORM;
    float*       negsum = w + W_NEG;
    const int gj = j0 + (lane & 15);
    const float ny = norms[gj];
    const int rbase = (lane >> 4) << 3;   // C/D layout: lanes>=16 hold M=8..15

    float l_dmax = -__builtin_inff();
    float l_maxi = -__builtin_inff();
    float l_mini =  __builtin_inff();

#pragma unroll
    for (int r = 0; r < 8; ++r) {
        const int gi = i0 + rbase + r;
        const float G  = c[r];
        const float nx = norms[gi];
        // c = 1:  A = 1-2G+ny, B = 1-nx, D = 1-2G+nx*ny
        const float Af = 1.f - 2.f * G + ny;
        const float Bf = 1.f - nx;
        const float Df = 1.f - 2.f * G + nx * ny;
        const float num2 = Af * Af * nx - 2.f * Af * Bf * G + Bf * Bf * ny;
        const float mob  = __builtin_sqrtf(fmaxf(num2, 0.f)) / fmaxf(Df, EPSF);
        const float u    = fminf(mob, CLIPF);          // mob >= 0
        const float dist = __logf((1.f + u) / (1.f - u));  // 2*artanh(u)
        const float sq   = dist * dist;

        const size_t idx = (size_t)gi * NPTS + gj;
        dist_out[idx] = sq;

        const bool pos = adj[idx] > 0;
        const float simi = fmaxf(__expf(-sq), EPSF);
        if (!pos) atomicAdd(&s_neg[rbase + r], simi);  // ds_add_f32

        l_dmax = fmaxf(l_dmax, sq);
        const float ei = pos ? G : 0.f;
        l_maxi = fmaxf(l_maxi, ei);
        l_mini = fminf(l_mini, ei);
    }

    s_dmax[threadIdx.x] = l_dmax;
    s_maxi[threadIdx.x] = l_maxi;
    s_mini[threadIdx.x] = l_mini;
    __syncthreads();

    for (int s = 128; s > 0; s >>= 1) {
        if (threadIdx.x < (unsigned)s) {
            s_dmax[threadIdx.x] = fmaxf(s_dmax[threadIdx.x], s_dmax[threadIdx.x + s]);
            s_maxi[threadIdx.x] = fmaxf(s_maxi[threadIdx.x], s_maxi[threadIdx.x + s]);
            s_mini[threadIdx.x] = fminf(s_mini[threadIdx.x], s_mini[threadIdx.x + s]);
        }
        __syncthreads();
    }

    if (threadIdx.x < 16) atomicAdd(&negsum[i0 + threadIdx.x], s_neg[threadIdx.x]);
    if (threadIdx.x == 0) {
        atomicMaxF(&w[W_CELL + 1], s_dmax[0]);
        atomicMaxF(&w[W_CELL + 2], s_maxi[0]);
        atomicMinF(&w[W_CELL + 3], s_mini[0]);
    }
}

// Kernel 2: loss = sum over edges of -log(max(simi/neg_sum_row, EPS)).
__global__ void __launch_bounds__(256)
loss_kernel(const float* __restrict__ dist_out,
            const int* __restrict__ adj,
            float* __restrict__ w) {
    __shared__ float s[256];
    const float* negsum = w + W_NEG;
    const size_t idx0 = (size_t)blockIdx.x * 256u * 8u + threadIdx.x;
    float acc = 0.f;
#pragma unroll
    for (int t = 0; t < 8; ++t) {
        const size_t idx = idx0 + (size_t)t * 256u;   // total = 8192*2048 = 4096^2
        if (adj[idx] > 0) {
            const int row = (int)(idx >> 12);
            const float sq   = dist_out[idx];
            const float simi = fmaxf(__expf(-sq), EPSF);
            const float ratio = fmaxf(simi / negsum[row], EPSF);
            acc -= __logf(ratio);
        }
    }
    s[threadIdx.x] = acc;
    __syncthreads();
    for (int st = 128; st > 0; st >>= 1) {
        if (threadIdx.x < (unsigned)st) s[threadIdx.x] += s[threadIdx.x + st];
        __syncthreads();
    }
    if (threadIdx.x == 0) atomicAdd(&w[W_CELL + 0], s[0]);
}

// Kernel 3: write the 4 scalars to the output tail.
__global__ void finalize_kernel(const float* __restrict__ w,
                                float* __restrict__ out_scal) {
    if (threadIdx.x == 0 && blockIdx.x == 0) {
        out_scal[0] = w[W_CELL + 0];   // loss
        out_scal[1] = w[W_CELL + 1];   // dist.max()
        out_scal[2] = w[W_CELL + 2];   // max_inner
        out_scal[3] = w[W_CELL + 3];   // min_inner
    }
}

extern "C" void kernel_launch(void* const* d_in, const int* in_sizes, int n_in,
                              void* d_out, int out_size, void* d_ws, size_t ws_size,
                              hipStream_t stream) {
    (void)in_sizes; (void)n_in; (void)out_size; (void)ws_size;
    const float* x   = (const float*)d_in[0];
    const int*   adj = (const int*)d_in[1];
    float* out = (float*)d_out;
    float* w   = (float*)d_ws;

    float* out_x    = out;                                   // 4096*16
    float* out_dist = out + (size_t)NPTS * DIMK;             // 4096*4096
    float* out_scal = out + (size_t)NPTS * DIMK + (size_t)NPTS * NPTS;  // 4 scalars

    init_kernel<<<NPTS / 256, 256, 0, stream>>>(x, out_x, w);

    dim3 g1(NPTS / 128, NPTS / 16);   // (32, 256): 8 waves/block, one 16x16 tile/wave
    gram_tile_kernel<<<g1, 256, 0, stream>>>(x, adj, out_dist, w);

    loss_kernel<<<(NPTS * (size_t)NPTS) / (256 * 8), 256, 0, stream>>>(out_dist, adj, w);

    finalize_kernel<<<1, 64, 0, stream>>>(w, out_scal);
}